// YOLOLoss_6622839571080
// MI455X (gfx1250) — compile-verified
//
#include <hip/hip_runtime.h>
#include <hip/hip_bf16.h>

// ---------------------------------------------------------------------------
// YOLO loss forward, MI455X (gfx1250, wave32).
//
// Roofline: reads 2 x 8192*7*7*30 f32 = 96.3 MB, writes 4 B. At 23.3 TB/s the
// kernel floor is ~4.1 us. Arithmetic is ~60 MFLOP total -> purely HBM bound,
// so the main loop streams with contiguous b64 loads (cell stride 120 B is
// 8-byte aligned) and all reduction work is noise. The 32-lane reduction is
// done on the matrix pipe with V_WMMA_F32_16X16X4_F32 (A = partials, B = ones)
// so the gfx1250 WMMA path is exercised where it is architecturally free.
// Two-pass reduction (no float atomics) => bitwise-deterministic output.
// ---------------------------------------------------------------------------

#define YB 8192
#define YS 7
#define YC 30
#define NCELLS (YB * YS * YS)   // 401408
#define BLOCK 256
#define NBLOCKS ((NCELLS + BLOCK - 1) / BLOCK)  // 1568

typedef float v2f __attribute__((ext_vector_type(2)));
typedef float v8f __attribute__((ext_vector_type(8)));

__device__ __forceinline__ float sq(float x) { return x * x; }

// Faithful replica of the reference calculate_iou, including the
// inter/area1 + area2 - inter quirk.
__device__ __forceinline__ float iou_quirk(const float* b1, const float* b2) {
    float area1     = b1[2] * b1[3];
    float area2     = b2[2] * b2[3];
    float max_left  = fmaxf(b1[0] - b1[2] * 0.5f, b2[0] - b2[2] * 0.5f);
    float min_right = fminf(b1[0] + b1[2] * 0.5f, b2[0] + b2[2] * 0.5f);
    float max_top   = fmaxf(b1[1] - b1[3] * 0.5f, b2[1] - b2[3] * 0.5f);
    float min_bot   = fminf(b1[1] + b1[3] * 0.5f, b2[1] + b2[3] * 0.5f);
    float inter     = (min_right - max_left) * (min_bot - max_top);
    bool  overlap   = (max_left < min_right) && (max_top < min_bot);
    return overlap ? (inter / area1 + area2 - inter) : 0.0f;
}

// Sum of one value per lane across all 32 lanes of the wave, using the matrix
// pipe. A (16x4): VGPR0 = x -> A[m][0] = x[m] (lanes 0-15), A[m][2] = x[m+16]
// (lanes 16-31); VGPR1 = 0 -> A[m][1] = A[m][3] = 0. B = ones(4x16).
// D[m][n] = x[m] + x[m+16] for every n. Lane (n<16) VGPR j holds row M=j,
// lane (n>=16) VGPR j holds row M=8+j. Summing the 8 D registers per lane and
// one shfl_xor(16) yields the full 32-lane total in every lane.
// Must be executed with EXEC == all ones (full wave).
__device__ __forceinline__ float wave32_reduce_wmma(float x) {
    v2f a; a[0] = x;    a[1] = 0.0f;
    v2f b; b[0] = 1.0f; b[1] = 1.0f;
    v8f c = {};
    v8f d = __builtin_amdgcn_wmma_f32_16x16x4_f32(
        /*neg_a=*/false, a, /*neg_b=*/false, b,
        /*c_mod=*/(short)0, c, /*reuse_a=*/false, /*reuse_b=*/false);
    float s = ((d[0] + d[1]) + (d[2] + d[3])) + ((d[4] + d[5]) + (d[6] + d[7]));
    s += __shfl_xor(s, 16, 32);   // combine the two half-wave row groups
    return s;
}

__global__ void __launch_bounds__(BLOCK)
yolo_partial(const float* __restrict__ preds,
             const float* __restrict__ labels,
             float* __restrict__ partials) {
    __shared__ float sdata[BLOCK];

    const int cell = blockIdx.x * BLOCK + threadIdx.x;
    float acc = 0.0f;

    if (cell < NCELLS) {
        const long long base = (long long)cell * YC;
        float p[YC], l[YC];
        // 120-byte cells are 8-byte aligned -> b64 streaming loads.
        const float2* pp = reinterpret_cast<const float2*>(preds  + base);
        const float2* lp = reinterpret_cast<const float2*>(labels + base);
#pragma unroll
        for (int i = 0; i < YC / 2; ++i) {
            float2 tp = pp[i];
            float2 tl = lp[i];
            p[2 * i] = tp.x; p[2 * i + 1] = tp.y;
            l[2 * i] = tl.x; l[2 * i + 1] = tl.y;
        }

        const float iou1 = iou_quirk(p + 0, l + 0);
        const float iou2 = iou_quirk(p + 5, l + 5);
        const bool  resp1 = iou1 > iou2;

        const float xy1 = sq(p[0] - l[0]) + sq(p[1] - l[1]);
        const float xy2 = sq(p[5] - l[5]) + sq(p[6] - l[6]);
        const float wh1 = sq(__builtin_sqrtf(p[2]) - __builtin_sqrtf(l[2])) +
                          sq(__builtin_sqrtf(p[3]) - __builtin_sqrtf(l[3]));
        const float wh2 = sq(__builtin_sqrtf(p[7]) - __builtin_sqrtf(l[7])) +
                          sq(__builtin_sqrtf(p[8]) - __builtin_sqrtf(l[8]));
        const float o1 = sq(p[4] - iou1);
        const float o2 = sq(p[9] - iou2);

        const float loss_xy    = 5.0f * (resp1 ? xy1 : xy2);
        const float loss_wh    = resp1 ? wh1 : wh2;
        const float loss_obj   = resp1 ? o1 : o2;
        const float loss_noobj = 0.5f * (resp1 ? o2 : o1);

        float loss_cls = 0.0f;
#pragma unroll
        for (int k = 10; k < YC; ++k) loss_cls += sq(p[k] - l[k]);

        const float obj_cell   = loss_xy + loss_wh + loss_obj + loss_noobj + loss_cls;
        const float noobj_cell = 0.5f * (p[4] * p[4] + p[9] * p[9]);

        acc = (l[4] == 1.0f) ? obj_cell : noobj_cell;
    }

    sdata[threadIdx.x] = acc;
    __syncthreads();

    // First wave (full 32 lanes, EXEC all ones) folds 256 -> 32 -> 1.
    if (threadIdx.x < 32) {
        float s = 0.0f;
#pragma unroll
        for (int j = 0; j < BLOCK / 32; ++j) s += sdata[threadIdx.x + 32 * j];
        s = wave32_reduce_wmma(s);
        if (threadIdx.x == 0) partials[blockIdx.x] = s;
    }
}

__global__ void __launch_bounds__(BLOCK)
yolo_finalize(const float* __restrict__ partials, int n, float* __restrict__ out) {
    __shared__ float sdata[BLOCK];

    float acc = 0.0f;
    for (int i = threadIdx.x; i < n; i += BLOCK) acc += partials[i];
    sdata[threadIdx.x] = acc;
    __syncthreads();

    if (threadIdx.x < 32) {
        float s = 0.0f;
#pragma unroll
        for (int j = 0; j < BLOCK / 32; ++j) s += sdata[threadIdx.x + 32 * j];
        s = wave32_reduce_wmma(s);
        if (threadIdx.x == 0) out[0] = s * (1.0f / (float)YB);
    }
}

extern "C" void kernel_launch(void* const* d_in, const int* in_sizes, int n_in,
                              void* d_out, int out_size, void* d_ws, size_t ws_size,
                              hipStream_t stream) {
    const float* preds  = (const float*)d_in[0];
    const float* labels = (const float*)d_in[1];
    float* out      = (float*)d_out;
    float* partials = (float*)d_ws;   // NBLOCKS floats of scratch

    yolo_partial<<<NBLOCKS, BLOCK, 0, stream>>>(preds, labels, partials);
    yolo_finalize<<<1, BLOCK, 0, stream>>>(partials, NBLOCKS, out);
}